// NodeLayer_1142461300897
// MI455X (gfx1250) — compile-verified
//
#include <hip/hip_runtime.h>
#include <math.h>

#define NN 50000
#define NE 500000
#define HD 128
#define BN_EPS 1e-5f

typedef __attribute__((ext_vector_type(2))) float v2f;
typedef __attribute__((ext_vector_type(8))) float v8f;

// Monotonic uint encoding of IEEE float so atomicMax(uint) == float max.
__device__ __forceinline__ unsigned fkey(float f) {
  unsigned u = __float_as_uint(f);
  return (u & 0x80000000u) ? ~u : (u | 0x80000000u);
}
__device__ __forceinline__ float funkey(unsigned k) {
  return (k & 0x80000000u) ? __uint_as_float(k & 0x7FFFFFFFu)
                           : __uint_as_float(~k);
}

__global__ void k_zero(float* __restrict__ p, int n) {
  int i = blockIdx.x * blockDim.x + threadIdx.x;
  int stride = gridDim.x * blockDim.x;
  for (; i < n; i += stride) p[i] = 0.0f;
}

// Pass 1: scores[e] = dot(emb[src], emb[dst]); segment-max into dst keys.
// One wave32 per edge; each lane loads one float4 (128 floats / 32 lanes).
__global__ void k_scores(const float* __restrict__ emb, const int* __restrict__ src,
                         const int* __restrict__ dst, float* __restrict__ scores,
                         unsigned* __restrict__ segmax) {
  int tid = threadIdx.x;
  int e = blockIdx.x * 8 + (tid >> 5);
  int lane = tid & 31;
  if (e >= NE) return;
  int s = src[e], d = dst[e];
  const float4* ps = (const float4*)(emb + (size_t)s * HD);
  const float4* pd = (const float4*)(emb + (size_t)d * HD);
  float4 a = ps[lane], b = pd[lane];
  float acc = a.x * b.x + a.y * b.y + a.z * b.z + a.w * b.w;
  #pragma unroll
  for (int off = 16; off > 0; off >>= 1) acc += __shfl_xor(acc, off, 32);
  if (lane == 0) {
    scores[e] = acc;
    atomicMax(&segmax[d], fkey(acc));
  }
}

// Pass 2: ex = exp(score - segmax[dst]); segment-sum into denom.
__global__ void k_exp(const int* __restrict__ dst, float* __restrict__ scores,
                      const unsigned* __restrict__ segmax, float* __restrict__ denom) {
  int e = blockIdx.x * blockDim.x + threadIdx.x;
  if (e >= NE) return;
  int d = dst[e];
  float m = funkey(segmax[d]);
  float ex = __expf(scores[e] - m);
  scores[e] = ex;
  atomicAdd(&denom[d], ex);
}

// Pass 3: neigh[dst] += emb[src] * (ex/denom[dst]); wave32 per edge, 4 atomics/lane.
__global__ void k_scatter(const float* __restrict__ emb, const int* __restrict__ src,
                          const int* __restrict__ dst, const float* __restrict__ ex,
                          const float* __restrict__ denom, float* __restrict__ neigh) {
  int tid = threadIdx.x;
  int e = blockIdx.x * 8 + (tid >> 5);
  int lane = tid & 31;
  if (e >= NE) return;
  int s = src[e], d = dst[e];
  float norm = ex[e] / denom[d];
  const float4* ps = (const float4*)(emb + (size_t)s * HD);
  float4 a = ps[lane];
  float* o = neigh + (size_t)d * HD + lane * 4;
  atomicAdd(o + 0, a.x * norm);
  atomicAdd(o + 1, a.y * norm);
  atomicAdd(o + 2, a.z * norm);
  atomicAdd(o + 3, a.w * norm);
}

// GEMM h = neigh @ W via V_WMMA_F32_16X16X4_F32 (exact f32 accumulation).
// 8 waves/block; wave w computes the 16x16 tile at (blockIdx*16, w*16).
// A 16x4 layout: lanes 0-15 hold K={0,1}, lanes 16-31 hold K={2,3}.
// B 4x16 mirrored; C/D: VGPR j -> row j (lanes 0-15) / j+8 (lanes 16-31).
__global__ void __launch_bounds__(256)
k_gemm_wmma(const float* __restrict__ neigh, const float* __restrict__ W,
            float* __restrict__ h) {
  int wave = threadIdx.x >> 5;
  int lane = threadIdx.x & 31;
  int hiHalf = lane >> 4;
  int lid = lane & 15;
  int rowBase = blockIdx.x * 16;
  int colBase = wave * 16;
  const float* arow = neigh + (size_t)(rowBase + lid) * HD;
  v8f c = {};
  #pragma unroll
  for (int k = 0; k < HD; k += 4) {
    int ka = k + 2 * hiHalf;
    v2f a;
    a.x = arow[ka];
    a.y = arow[ka + 1];
    v2f b;
    b.x = W[(size_t)ka * HD + colBase + lid];
    b.y = W[(size_t)(ka + 1) * HD + colBase + lid];
    c = __builtin_amdgcn_wmma_f32_16x16x4_f32(false, a, false, b,
                                              (short)0, c, false, false);
  }
  #pragma unroll
  for (int j = 0; j < 8; ++j)
    h[(size_t)(rowBase + j + 8 * hiHalf) * HD + colBase + lid] = c[j];
}

// Column sums / sums of squares for BatchNorm batch stats.
__global__ void k_stats(const float* __restrict__ h, float* __restrict__ colsum,
                        float* __restrict__ colsq) {
  __shared__ float ssum[256], ssq[256];
  int tid = threadIdx.x;
  int col = tid & 127;
  int rphase = tid >> 7;  // 0 or 1: two rows per block per iteration
  float s = 0.f, q = 0.f;
  for (int r = blockIdx.x * 2 + rphase; r < NN; r += gridDim.x * 2) {
    float v = h[(size_t)r * HD + col];
    s += v;
    q += v * v;
  }
  ssum[tid] = s;
  ssq[tid] = q;
  __syncthreads();
  if (tid < 128) {
    atomicAdd(&colsum[col], ssum[tid] + ssum[tid + 128]);
    atomicAdd(&colsq[col], ssq[tid] + ssq[tid + 128]);
  }
}

// BN (batch stats, biased var) + tanh, in place on d_out.
__global__ void k_bn_tanh(float* __restrict__ h, const float* __restrict__ colsum,
                          const float* __restrict__ colsq,
                          const float* __restrict__ gamma,
                          const float* __restrict__ beta) {
  int i = blockIdx.x * blockDim.x + threadIdx.x;
  if (i >= NN * HD) return;
  int col = i & 127;
  const float invN = 1.0f / (float)NN;
  float mu = colsum[col] * invN;
  float var = colsq[col] * invN - mu * mu;
  float y = (h[i] - mu) * rsqrtf(var + BN_EPS) * gamma[col] + beta[col];
  h[i] = tanhf(y);
}

extern "C" void kernel_launch(void* const* d_in, const int* in_sizes, int n_in,
                              void* d_out, int out_size, void* d_ws, size_t ws_size,
                              hipStream_t stream) {
  const float* emb = (const float*)d_in[0];   // [50000,128]
  const float* W = (const float*)d_in[1];     // [128,128]
  const float* gamma = (const float*)d_in[2]; // [128]
  const float* beta = (const float*)d_in[3];  // [128]
  const int* src = (const int*)d_in[4];       // [500000]
  const int* dst = (const int*)d_in[5];       // [500000]
  float* out = (float*)d_out;                 // [50000,128]

  float* ws = (float*)d_ws;
  float* scores = ws;                           // [E]      offset 0
  unsigned* segmax = (unsigned*)(ws + 500000);  // [N]      keys
  float* denom = ws + 550000;                   // [N]
  float* neigh = ws + 600000;                   // [N*128]  -> ends at 7,000,000
  float* colsum = ws + 7000000;                 // [128]
  float* colsq = ws + 7000128;                  // [128]

  // Re-zero all reduction buffers every call (deterministic across replays).
  k_zero<<<1024, 256, 0, stream>>>(ws + 500000, 7000256 - 500000);

  k_scores<<<(NE + 7) / 8, 256, 0, stream>>>(emb, src, dst, scores, segmax);
  k_exp<<<(NE + 255) / 256, 256, 0, stream>>>(dst, scores, segmax, denom);
  k_scatter<<<(NE + 7) / 8, 256, 0, stream>>>(emb, src, dst, scores, denom, neigh);

  k_gemm_wmma<<<NN / 16, 256, 0, stream>>>(neigh, W, out);

  k_stats<<<128, 256, 0, stream>>>(out, colsum, colsq);
  k_bn_tanh<<<(NN * HD + 255) / 256, 256, 0, stream>>>(out, colsum, colsq, gamma, beta);
}